// BSplineActivation_30691836297563
// MI455X (gfx1250) — compile-verified
//
#include <hip/hip_runtime.h>
#include <hip/hip_bf16.h>
#include <stdint.h>

// ---------------------------------------------------------------------------
// KAN B-spline activation for MI455X (gfx1250), compile-only tuned.
//
// Shapes: x (8192,1024) f32.  Per row: layernorm -> uniform cubic B-spline
// (grid_size=5, order=3 -> 8 coeffs/feature) -> + base_weight*silu(normed).
//
// Roofline: 64 MB moved -> ~2.7us @ 23.3 TB/s.  Closed-form cardinal cubic
// B-spline (~45 f32 ops/elem) instead of full Cox-de Boor (~500) keeps us
// near the memory roof.  WMMA is *not* used: the einsum 'bfk,fk->bf' is a
// per-feature length-8 dot (diagonal), a 16x16 matrix op would burn 16x
// redundant columns with no reuse.
//
// CDNA5-specific paths used:
//   * TENSOR_LOAD_TO_LDS (TDM): 4 descriptors stage all tables into LDS;
//     the spline-weight copy uses TDM's pad_enable to emit a stride-9
//     (bank-conflict-padded) LDS layout directly from the DMA engine.
//   * s_wait_tensorcnt for TDM completion.
//   * wave32 __shfl_xor reduction (one wave per row, no LDS reduction).
//   * global_prefetch for the next row of x.
// ---------------------------------------------------------------------------

typedef unsigned int v4u __attribute__((ext_vector_type(4)));
typedef int          v4i __attribute__((ext_vector_type(4)));
typedef int          v8i __attribute__((ext_vector_type(8)));

#define NFEAT    1024
#define NCOEF    8          // grid_size + spline_order
#define WSTRIDE  9          // padded LDS stride (dwords) for spline weights
#define NKNOT    12         // grid_size + 2*spline_order + 1
#define WAVES    8          // waves per block (256 threads, wave32)

// 1-D TDM copy: global -> LDS, ndw dwords, optional LDS padding of
// pad_amount+1 dwords every (2<<pad_interval) dwords.
__device__ __forceinline__ void tdm_load_1d(const void* gptr, uint32_t lds_byte,
                                            uint32_t ndw, uint32_t pad_en,
                                            uint32_t pad_interval, uint32_t pad_amount) {
  uint64_t ga = (uint64_t)(size_t)gptr;

  // D# group 0: count=1 | lds_addr | global_addr[56:0] | type=2
  v4u g0;
  g0.x = 1u;                                                   // count = 1 (valid)
  g0.y = lds_byte;                                             // lds_addr (bytes)
  g0.z = (uint32_t)ga;                                         // global_addr[31:0]
  g0.w = (uint32_t)((ga >> 32) & 0x01FFFFFFu) | 0x80000000u;   // addr[56:32] | type=2

  // D# group 1: data_size=4B, 1-D tile of ndw elements.
  uint32_t w0 = (2u << 16) | (pad_en << 20) | (pad_interval << 22) | (pad_amount << 25);
  uint32_t w1 = (ndw & 0xFFFFu) << 16;            // tensor_dim0[15:0]  (abar=0)
  uint32_t w2 = (ndw >> 16) | (1u << 16);         // tensor_dim0[31:16] | tensor_dim1=1
  uint32_t w3 = (ndw & 0xFFFFu) << 16;            // tile_dim0 (<=65535)
  uint32_t w4 = 0;                                // tile_dim1=0, tile_dim2=0 (1-D)
  uint32_t w5 = ndw;                              // tensor_dim0_stride[31:0]
  v8i g1 = { (int)w0, (int)w1, (int)w2, (int)w3, (int)w4, (int)w5, 0, 0 };

  v4i z4 = {0, 0, 0, 0};
#if defined(__clang_major__) && (__clang_major__ >= 23)
  v8i z8 = {0, 0, 0, 0, 0, 0, 0, 0};
  __builtin_amdgcn_tensor_load_to_lds(g0, g1, z4, z4, z8, 0);
#else
  __builtin_amdgcn_tensor_load_to_lds(g0, g1, z4, z4, 0);
#endif
}

// Per-element: layernorm'd value -> spline dot + silu residual.
__device__ __forceinline__ float kan_elem(float xv, float mu, float rs,
                                          float lw, float lb, float bwv,
                                          int f, const float* __restrict__ s_w,
                                          float inv_h, float c0) {
  float n  = fmaf((xv - mu) * rs, lw, lb);        // layernorm
  float u  = fmaf(n, inv_h, c0);                  // knot-space coordinate
  float fi = floorf(u);
  int   i  = (int)fi;
  float t  = u - fi;                              // fraction within interval
  bool  inr = (u >= 0.0f) && (i <= 10);           // outside grid -> spline = 0

  // Cardinal cubic B-spline weights for bases j = i-3 .. i
  float t2 = t * t, t3 = t2 * t;
  float s1 = 1.0f - t;
  float B0 = s1 * s1 * s1 * (1.0f / 6.0f);                       // j = i-3
  float B1 = fmaf(0.5f, t3, fmaf(-1.0f, t2, 2.0f / 3.0f));       // j = i-2
  float B3 = t3 * (1.0f / 6.0f);                                 // j = i
  float B2 = 1.0f - B0 - B1 - B3;                                // j = i-1
  if (!inr) { B0 = B1 = B2 = B3 = 0.0f; }

  int ii  = min(max(i, 0), 10);
  int jlo = ii - 3;
  const float* wrow = s_w + f * WSTRIDE;
  float acc = 0.0f;
#pragma unroll
  for (int tt = 0; tt < 4; ++tt) {
    int   jj = jlo + tt;
    int   jc = min(max(jj, 0), NCOEF - 1);
    float wv = wrow[jc];                          // LDS gather (stride-9 padded)
    float Bt = (tt == 0) ? B0 : (tt == 1) ? B1 : (tt == 2) ? B2 : B3;
    Bt = (jj == jc) ? Bt : 0.0f;                  // drop bases outside [0,7]
    acc = fmaf(Bt, wv, acc);
  }

  // silu residual
  float sig = __builtin_amdgcn_rcpf(1.0f + __expf(-n));
  return fmaf(bwv, n * sig, acc);
}

__global__ __launch_bounds__(32 * WAVES) void
kan_bspline_kernel(const float* __restrict__ x,
                   const float* __restrict__ ln_w,
                   const float* __restrict__ ln_b,
                   const float* __restrict__ sw,
                   const float* __restrict__ bw,
                   const float* __restrict__ grid,
                   float* __restrict__ out, int rows) {
  // 48 KB LDS: padded spline weights + ln_w + ln_b + base_w
  __shared__ __align__(16) float smem[NFEAT * WSTRIDE + 3 * NFEAT];
  float* s_w  = smem;                        // [NFEAT*9]
  float* s_lw = smem + NFEAT * WSTRIDE;      // [NFEAT]
  float* s_lb = s_lw + NFEAT;                // [NFEAT]
  float* s_bw = s_lb + NFEAT;                // [NFEAT]

  // --- Stage all tables with the Tensor Data Mover (wave 0 only). ---------
  if (threadIdx.x < 32u) {
    uint32_t base = (uint32_t)(size_t)(void*)smem;   // LDS byte offset (addr[31:0])
    // spline_weight: 1024x8 dwords, pad 1 dword every 8 -> stride-9 layout
    tdm_load_1d(sw,   base,                               NFEAT * NCOEF, 1u, 2u, 0u);
    tdm_load_1d(ln_w, base + NFEAT * WSTRIDE * 4u,        NFEAT,         0u, 0u, 0u);
    tdm_load_1d(ln_b, base + (NFEAT * WSTRIDE + NFEAT) * 4u,     NFEAT,  0u, 0u, 0u);
    tdm_load_1d(bw,   base + (NFEAT * WSTRIDE + 2 * NFEAT) * 4u, NFEAT,  0u, 0u, 0u);
    __builtin_amdgcn_s_wait_tensorcnt(0);
  }
  __syncthreads();

  const int lane = (int)(threadIdx.x & 31u);
  const int wv   = (int)(threadIdx.x >> 5);

  // Grid is a broadcast uniform linspace: derive knot origin/step once.
  const float gk0   = grid[0];
  const float hstep = grid[1] - grid[0];
  const float inv_h = 1.0f / hstep;
  const float c0    = -gk0 * inv_h;

  const int rstride = (int)gridDim.x * WAVES;
  for (int row = (int)blockIdx.x * WAVES + wv; row < rows; row += rstride) {
    const float4* __restrict__ xr = (const float4*)(x + (size_t)row * NFEAT);
    float4* __restrict__ orow     = (float4*)(out + (size_t)row * NFEAT);

    // Prefetch next row assigned to this wave.
    int nrow = row + rstride;
    if (nrow < rows) __builtin_prefetch(x + (size_t)nrow * NFEAT + (size_t)lane * 8, 0, 3);

    // Load 32 features/lane as 8 x float4 (coalesced b128), keep in VGPRs.
    float4 v[8];
#pragma unroll
    for (int j = 0; j < 8; ++j) v[j] = xr[lane + 32 * j];

    // Row mean / variance: per-lane accumulate + wave32 xor-shuffle tree.
    float s = 0.0f, ss = 0.0f;
#pragma unroll
    for (int j = 0; j < 8; ++j) {
      s  += v[j].x + v[j].y + v[j].z + v[j].w;
      ss += v[j].x * v[j].x + v[j].y * v[j].y + v[j].z * v[j].z + v[j].w * v[j].w;
    }
#pragma unroll
    for (int off = 16; off > 0; off >>= 1) {
      s  += __shfl_xor(s, off, 32);
      ss += __shfl_xor(ss, off, 32);
    }
    const float mu = s * (1.0f / NFEAT);
    const float var = ss * (1.0f / NFEAT) - mu * mu;
    const float rs = __builtin_amdgcn_rsqf(var + 1e-5f);

    // Evaluate 32 features/lane, write back as float4.
#pragma unroll
    for (int j = 0; j < 8; ++j) {
      const int f4 = lane + 32 * j;
      const float4 lw  = ((const float4*)s_lw)[f4];
      const float4 lb  = ((const float4*)s_lb)[f4];
      const float4 bwq = ((const float4*)s_bw)[f4];
      const int f = f4 * 4;
      float4 o;
      o.x = kan_elem(v[j].x, mu, rs, lw.x, lb.x, bwq.x, f + 0, s_w, inv_h, c0);
      o.y = kan_elem(v[j].y, mu, rs, lw.y, lb.y, bwq.y, f + 1, s_w, inv_h, c0);
      o.z = kan_elem(v[j].z, mu, rs, lw.z, lb.z, bwq.z, f + 2, s_w, inv_h, c0);
      o.w = kan_elem(v[j].w, mu, rs, lw.w, lb.w, bwq.w, f + 3, s_w, inv_h, c0);
      orow[f4] = o;
    }
  }
}

extern "C" void kernel_launch(void* const* d_in, const int* in_sizes, int n_in,
                              void* d_out, int out_size, void* d_ws, size_t ws_size,
                              hipStream_t stream) {
  const float* x    = (const float*)d_in[0];  // (rows, 1024)
  const float* lnw  = (const float*)d_in[1];  // (1024,)
  const float* lnb  = (const float*)d_in[2];  // (1024,)
  const float* sw   = (const float*)d_in[3];  // (1024, 8)
  const float* bw   = (const float*)d_in[4];  // (1024,)
  const float* grid = (const float*)d_in[5];  // (1024, 12) broadcast linspace
  float* out = (float*)d_out;

  const int rows = in_sizes[0] / NFEAT;
  int blocks = 256;                               // amortize TDM staging over ~32 rows/block
  const int need = (rows + WAVES - 1) / WAVES;
  if (blocks > need) blocks = need;
  if (blocks < 1) blocks = 1;

  hipLaunchKernelGGL(kan_bspline_kernel, dim3(blocks), dim3(32 * WAVES), 0, stream,
                     x, lnw, lnb, sw, bw, grid, out, rows);
}